// GNNPortScoreFaultAware_53283364274536
// MI455X (gfx1250) — compile-verified
//
#include <hip/hip_runtime.h>
#include <cfloat>

// ---------------------------------------------------------------------------
// GNN port-score pipeline for MI455X (gfx1250, wave32).
//  * All dense linears (GAT lin_l/lin_r, decoder W1 halves) run through
//    V_WMMA_F32_16X16X4_F32 (native f32 WMMA), one wave per 16x16 output tile.
//  * All-pairs decoder: pairs@W1 is split algebraically into per-node GEMMs
//    A[i]=emb@W1[:32]+b1, B[j]=emb@W1[32:]; the 1M-pair kernel then does
//    LN+LReLU+dot from LDS tiles (elementwise => VALU by necessity).
//  * Decoder tile staging uses GLOBAL_LOAD_ASYNC_TO_LDS (ASYNCcnt) when the
//    toolchain exposes the builtin; falls back to regular loads otherwise.
//  * Segment softmax via ordered-uint atomicMax + f32 atomicAdd.
// ---------------------------------------------------------------------------

#define NN   1024
#define EE   8192
#define ETOT 9216   // EE + NN self loops

#if defined(__HIP_DEVICE_COMPILE__) && __has_builtin(__builtin_amdgcn_global_load_async_to_lds_b32)
#define USE_ASYNC_LDS 1
typedef __attribute__((address_space(1))) int g_i32;   // global int*
typedef __attribute__((address_space(3))) int l_i32;   // LDS int*
#endif

typedef __attribute__((ext_vector_type(2))) float v2f;
typedef __attribute__((ext_vector_type(8))) float v8f;

__device__ __forceinline__ unsigned ford(float f) {
  unsigned u = __float_as_uint(f);
  return (u & 0x80000000u) ? ~u : (u | 0x80000000u);
}
__device__ __forceinline__ float fdec(unsigned u) {
  unsigned v = (u & 0x80000000u) ? (u ^ 0x80000000u) : ~u;
  return __uint_as_float(v);
}

// ---------------- mean of edge_attr (fill_value='mean') ---------------------
__global__ void edge_mean_kernel(const float* __restrict__ ea, float* __restrict__ out) {
  __shared__ float s[256];
  float acc = 0.f;
  for (int i = threadIdx.x; i < EE; i += 256) acc += ea[i];
  s[threadIdx.x] = acc;
  __syncthreads();
  for (int st = 128; st > 0; st >>= 1) {
    if (threadIdx.x < st) s[threadIdx.x] += s[threadIdx.x + st];
    __syncthreads();
  }
  if (threadIdx.x == 0) out[0] = s[0] / (float)EE;
}

// ---------------- f32 WMMA GEMM: out[M,Nc] = X[M,K] @ W[K,Nc] + bias --------
// One wave (32 threads) computes one 16x16 tile via V_WMMA_F32_16X16X4_F32.
// A frag  : lane%16 = M row, lane/16 selects K pair {0,1}|{2,3}, v2f per lane.
// B frag  : lane%16 = N col, same K-pair split.
// C/D frag: v8f; VGPR v -> M = v + 8*(lane>=16), N = lane%16.
__global__ void linear_wmma_kernel(const float* __restrict__ X,
                                   const float* __restrict__ W,
                                   const float* __restrict__ bias,
                                   float* __restrict__ out,
                                   int K, int ldX, int Nc) {
  const int lane = threadIdx.x;        // 0..31, all active (EXEC must be ~0)
  const int n0 = blockIdx.x * 16;
  const int m0 = blockIdx.y * 16;
  const int half = lane >> 4;          // 0 | 1
  const int l15  = lane & 15;
  const float* xrow = X + (size_t)(m0 + l15) * ldX + half * 2;
  const float* wcol = W + (size_t)half * 2 * Nc + n0 + l15;
  v8f acc = {0.f, 0.f, 0.f, 0.f, 0.f, 0.f, 0.f, 0.f};
  for (int kb = 0; kb < K; kb += 4) {
    v2f a, b;
    a.x = xrow[kb];
    a.y = xrow[kb + 1];
    b.x = wcol[(size_t)kb * Nc];
    b.y = wcol[(size_t)(kb + 1) * Nc];
    acc = __builtin_amdgcn_wmma_f32_16x16x4_f32(false, a, false, b,
                                                (short)0, acc, false, false);
  }
  const float bb = bias ? bias[n0 + l15] : 0.f;
  const int rbase = m0 + half * 8;
#pragma unroll
  for (int v = 0; v < 8; ++v)
    out[(size_t)(rbase + v) * Nc + n0 + l15] = acc[v] + bb;
}

// ---------------- per-layer segment buffers init ----------------------------
__global__ void init_seg_kernel(unsigned* __restrict__ amax, float* __restrict__ denom,
                                float* __restrict__ agg, int nh, int nhd) {
  const int i = blockIdx.x * blockDim.x + threadIdx.x;
  if (i < nh) { amax[i] = ford(-FLT_MAX); denom[i] = 0.f; }
  if (i < nhd) agg[i] = 0.f;
}

__device__ __forceinline__ void edge_src_dst(const int* __restrict__ ei, int e,
                                             int& src, int& dst) {
  if (e < EE) { src = ei[e]; dst = ei[EE + e]; }
  else        { src = dst = e - EE; }
}

// ---------------- GATv2 edge scoring (alpha) + segment max ------------------
__global__ void edge_alpha_kernel(const float* __restrict__ xl, const float* __restrict__ xr,
                                  const int* __restrict__ ei, const float* __restrict__ eattr,
                                  const float* __restrict__ ea_mean,
                                  const float* __restrict__ We, const float* __restrict__ att,
                                  float* __restrict__ alpha, unsigned* __restrict__ amax,
                                  int H, int C) {
  const int t = blockIdx.x * blockDim.x + threadIdx.x;
  if (t >= ETOT * H) return;
  const int e = t / H, h = t - e * H;
  int src, dst;
  edge_src_dst(ei, e, src, dst);
  const float ea = (e < EE) ? eattr[e] : ea_mean[0];
  const int HD = H * C;
  const float* pl = xl + (size_t)src * HD + h * C;
  const float* pr = xr + (size_t)dst * HD + h * C;
  const float* pw = We + h * C;
  const float* pa = att + h * C;
  float a = 0.f;
  for (int c = 0; c < C; ++c) {
    float m = pl[c] + pr[c] + ea * pw[c];
    m = (m > 0.f) ? m : 0.2f * m;         // LeakyReLU(0.2)
    a += m * pa[c];
  }
  alpha[t] = a;
  atomicMax(&amax[dst * H + h], ford(a));
}

// ---------------- exp(alpha - max) and segment sum --------------------------
__global__ void edge_expsum_kernel(const int* __restrict__ ei,
                                   const float* __restrict__ alpha,
                                   const unsigned* __restrict__ amax,
                                   float* __restrict__ expv, float* __restrict__ denom,
                                   int H) {
  const int t = blockIdx.x * blockDim.x + threadIdx.x;
  if (t >= ETOT * H) return;
  const int e = t / H, h = t - e * H;
  int src, dst;
  edge_src_dst(ei, e, src, dst);
  (void)src;
  const float ev = expf(alpha[t] - fdec(amax[dst * H + h]));
  expv[t] = ev;
  atomicAdd(&denom[dst * H + h], ev);
}

// ---------------- weighted message aggregation ------------------------------
__global__ void edge_agg_kernel(const float* __restrict__ xl, const int* __restrict__ ei,
                                const float* __restrict__ expv, const float* __restrict__ denom,
                                float* __restrict__ agg, int H, int C) {
  const int HD = H * C;
  const int t = blockIdx.x * blockDim.x + threadIdx.x;
  if (t >= ETOT * HD) return;
  const int e = t / HD, r = t - e * HD;
  const int h = r / C;
  int src, dst;
  edge_src_dst(ei, e, src, dst);
  const float w = expv[e * H + h] / (denom[dst * H + h] + 1e-16f);
  atomicAdd(&agg[(size_t)dst * HD + r], xl[(size_t)src * HD + r] * w);
}

// ---------------- per-node bias + LayerNorm (+ ELU) -------------------------
__global__ void node_ln_kernel(const float* __restrict__ agg, const float* __restrict__ bias,
                               const float* __restrict__ g, const float* __restrict__ b,
                               float* __restrict__ out, float* __restrict__ out2,
                               int HD, int do_elu) {
  __shared__ float s[64];
  const int n = blockIdx.x, t = threadIdx.x;
  const float v = agg[(size_t)n * HD + t] + bias[t];
  s[t] = v;
  __syncthreads();
  for (int st = HD >> 1; st > 0; st >>= 1) {
    if (t < st) s[t] += s[t + st];
    __syncthreads();
  }
  const float mu = s[0] / (float)HD;
  __syncthreads();
  const float d = v - mu;
  s[t] = d * d;
  __syncthreads();
  for (int st = HD >> 1; st > 0; st >>= 1) {
    if (t < st) s[t] += s[t + st];
    __syncthreads();
  }
  const float var = s[0] / (float)HD;
  float y = d * rsqrtf(var + 1e-5f) * g[t] + b[t];
  if (do_elu) y = (y > 0.f) ? y : expm1f(y);   // ELU(alpha=1)
  out[(size_t)n * HD + t] = y;
  if (out2) out2[(size_t)n * HD + t] = y;
}

// ---------------- all-pairs decoder -----------------------------------------
struct DecP {
  const float* g[4];
  const float* bn[4];
  const float* w2[4];
  const float* b2[4];
};

__global__ __launch_bounds__(256) void pair_decoder_kernel(
    const float* __restrict__ A, const float* __restrict__ B, DecP p,
    float* __restrict__ scores) {
  __shared__ float As[4][16][33];   // padded: stride 33 kills bank conflicts
  __shared__ float Bs[4][16][33];
  __shared__ float Gs[4][32], Bns[4][32], W2s[4][32], b2s[4];
  const int tid = threadIdx.y * 16 + threadIdx.x;
  const int i0 = blockIdx.y * 16, j0 = blockIdx.x * 16;

#if defined(USE_ASYNC_LDS)
  // Stage A/B tiles LDS-direct via the gfx1250 async global->LDS path
  // (tracked by ASYNCcnt, no VGPR round-trip). b32 beats because the padded
  // LDS row stride (132B) is only 4B-aligned.
#pragma unroll
  for (int it = 0; it < 8; ++it) {
    const int idx = tid + it * 256;            // 0..2047 = 4 dec * 16 rows * 32 ch
    const int d = idx >> 9, rem = idx & 511, row = rem >> 5, k = rem & 31;
    __builtin_amdgcn_global_load_async_to_lds_b32(
        (g_i32*)(A + (size_t)d * NN * 32 + (size_t)(i0 + row) * 32 + k),
        (l_i32*)&As[d][row][k], 0, 0);
    __builtin_amdgcn_global_load_async_to_lds_b32(
        (g_i32*)(B + (size_t)d * NN * 32 + (size_t)(j0 + row) * 32 + k),
        (l_i32*)&Bs[d][row][k], 0, 0);
  }
#else
#pragma unroll
  for (int it = 0; it < 8; ++it) {
    const int idx = tid + it * 256;
    const int d = idx >> 9, rem = idx & 511, row = rem >> 5, k = rem & 31;
    As[d][row][k] = A[(size_t)d * NN * 32 + (size_t)(i0 + row) * 32 + k];
    Bs[d][row][k] = B[(size_t)d * NN * 32 + (size_t)(j0 + row) * 32 + k];
  }
#endif

  if (tid < 128) {
    const int d = tid >> 5, k = tid & 31;
    Gs[d][k]  = p.g[d][k];
    Bns[d][k] = p.bn[d][k];
    W2s[d][k] = p.w2[d][k];
  }
  if (tid < 4) b2s[tid] = p.b2[tid][0];

#if defined(USE_ASYNC_LDS)
#if __has_builtin(__builtin_amdgcn_s_wait_asynccnt)
  __builtin_amdgcn_s_wait_asynccnt(0);
#else
  asm volatile("s_wait_asynccnt 0x0" ::: "memory");
#endif
#endif
  __syncthreads();

  const int ti = threadIdx.y, tj = threadIdx.x;
  float res[4];
#pragma unroll
  for (int d = 0; d < 4; ++d) {
    float hbuf[32];
    float s = 0.f, s2 = 0.f;
#pragma unroll
    for (int k = 0; k < 32; ++k) {
      const float h = As[d][ti][k] + Bs[d][tj][k];
      hbuf[k] = h;
      s += h;
      s2 += h * h;
    }
    const float mu = s * (1.f / 32.f);
    const float var = s2 * (1.f / 32.f) - mu * mu;
    const float rs = rsqrtf(var + 1e-5f);
    float acc = 0.f;
#pragma unroll
    for (int k = 0; k < 32; ++k) {
      float y = (hbuf[k] - mu) * rs * Gs[d][k] + Bns[d][k];
      y = (y > 0.f) ? y : 0.1f * y;            // LeakyReLU(0.1)
      acc += y * W2s[d][k];
    }
    res[d] = acc + b2s[d];
  }
  const size_t i = (size_t)(i0 + ti), j = (size_t)(j0 + tj);
  *reinterpret_cast<float4*>(scores + (i * NN + j) * 4) =
      make_float4(res[0], res[1], res[2], res[3]);
}

// ---------------------------------------------------------------------------
static void run_gat_layer(const float* X, int D, int H, int C,
                          const float* Wl, const float* bl,
                          const float* Wr, const float* br,
                          const float* We, const float* att, const float* bias,
                          const float* lng, const float* lnb, int do_elu,
                          const int* ei, const float* eattr, const float* ea_mean,
                          float* xl, float* xr, float* alpha, float* expv,
                          unsigned* amax, float* denom, float* agg,
                          float* out, float* out2, hipStream_t stream) {
  const int HD = H * C;
  dim3 gg(HD / 16, NN / 16);
  linear_wmma_kernel<<<gg, 32, 0, stream>>>(X, Wl, bl, xl, D, D, HD);
  linear_wmma_kernel<<<gg, 32, 0, stream>>>(X, Wr, br, xr, D, D, HD);
  const int nh = NN * H, nhd = NN * HD;
  init_seg_kernel<<<(nhd + 255) / 256, 256, 0, stream>>>(amax, denom, agg, nh, nhd);
  edge_alpha_kernel<<<(ETOT * H + 255) / 256, 256, 0, stream>>>(
      xl, xr, ei, eattr, ea_mean, We, att, alpha, amax, H, C);
  edge_expsum_kernel<<<(ETOT * H + 255) / 256, 256, 0, stream>>>(
      ei, alpha, amax, expv, denom, H);
  edge_agg_kernel<<<(ETOT * HD + 255) / 256, 256, 0, stream>>>(
      xl, ei, expv, denom, agg, H, C);
  node_ln_kernel<<<NN, HD, 0, stream>>>(agg, bias, lng, lnb, out, out2, HD, do_elu);
}

extern "C" void kernel_launch(void* const* d_in, const int* in_sizes, int n_in,
                              void* d_out, int out_size, void* d_ws, size_t ws_size,
                              hipStream_t stream) {
  (void)in_sizes; (void)n_in; (void)out_size; (void)ws_size;
  // Input order: x, edge_index, edge_attr, then params tree-flattened
  // (dict keys sorted): c1{We,Wl,Wr,att,bias,bl,br} c2{...} c3{...}
  // dec[0..3]{W1,W2,b1,b2,bn,g} n1(g,b) n2(g,b) n3(g,b)
  const float* x     = (const float*)d_in[0];
  const int*   ei    = (const int*)d_in[1];
  const float* eattr = (const float*)d_in[2];
  const float* We1 = (const float*)d_in[3],  *Wl1 = (const float*)d_in[4],
             * Wr1 = (const float*)d_in[5],  *att1 = (const float*)d_in[6],
             * bias1 = (const float*)d_in[7], *bl1 = (const float*)d_in[8],
             * br1 = (const float*)d_in[9];
  const float* We2 = (const float*)d_in[10], *Wl2 = (const float*)d_in[11],
             * Wr2 = (const float*)d_in[12], *att2 = (const float*)d_in[13],
             * bias2 = (const float*)d_in[14], *bl2 = (const float*)d_in[15],
             * br2 = (const float*)d_in[16];
  const float* We3 = (const float*)d_in[17], *Wl3 = (const float*)d_in[18],
             * Wr3 = (const float*)d_in[19], *att3 = (const float*)d_in[20],
             * bias3 = (const float*)d_in[21], *bl3 = (const float*)d_in[22],
             * br3 = (const float*)d_in[23];
  const float* n1g = (const float*)d_in[48], *n1b = (const float*)d_in[49];
  const float* n2g = (const float*)d_in[50], *n2b = (const float*)d_in[51];
  const float* n3g = (const float*)d_in[52], *n3b = (const float*)d_in[53];

  float* scores = (float*)d_out;                       // [N,N,4]
  float* emb_out = scores + (size_t)NN * NN * 4;       // [N,32]

  // workspace layout (floats)
  float* w = (float*)d_ws;
  float* ea_mean = w;                 w += 16;
  float* xl      = w;                 w += (size_t)NN * 64;
  float* xr      = w;                 w += (size_t)NN * 64;
  float* alpha   = w;                 w += (size_t)ETOT * 4;
  float* expv    = w;                 w += (size_t)ETOT * 4;
  unsigned* amax = (unsigned*)w;      w += (size_t)NN * 4;
  float* denom   = w;                 w += (size_t)NN * 4;
  float* agg     = w;                 w += (size_t)NN * 64;
  float* h1      = w;                 w += (size_t)NN * 64;
  float* h2      = w;                 w += (size_t)NN * 64;
  float* emb     = w;                 w += (size_t)NN * 32;
  float* Abuf    = w;                 w += (size_t)4 * NN * 32;
  float* Bbuf    = w;                 w += (size_t)4 * NN * 32;

  edge_mean_kernel<<<1, 256, 0, stream>>>(eattr, ea_mean);

  // GAT layer 1: in 12 -> 4 heads x 16, LN + ELU
  run_gat_layer(x, 12, 4, 16, Wl1, bl1, Wr1, br1, We1, att1, bias1, n1g, n1b, 1,
                ei, eattr, ea_mean, xl, xr, alpha, expv, amax, denom, agg,
                h1, nullptr, stream);
  // GAT layer 2: 64 -> 4 x 16, LN + ELU
  run_gat_layer(h1, 64, 4, 16, Wl2, bl2, Wr2, br2, We2, att2, bias2, n2g, n2b, 1,
                ei, eattr, ea_mean, xl, xr, alpha, expv, amax, denom, agg,
                h2, nullptr, stream);
  // GAT layer 3: 64 -> 1 x 32, LN only; dual-write emb to d_out tail
  run_gat_layer(h2, 64, 1, 32, Wl3, bl3, Wr3, br3, We3, att3, bias3, n3g, n3b, 0,
                ei, eattr, ea_mean, xl, xr, alpha, expv, amax, denom, agg,
                emb, emb_out, stream);

  // decoder precompute: A_d = emb @ W1[:32] + b1 ; B_d = emb @ W1[32:]
  DecP p;
  for (int d = 0; d < 4; ++d) {
    const int base = 24 + 6 * d;
    const float* W1 = (const float*)d_in[base + 0];    // [64,32]
    const float* W2 = (const float*)d_in[base + 1];    // [32]
    const float* b1 = (const float*)d_in[base + 2];
    const float* b2 = (const float*)d_in[base + 3];
    const float* bn = (const float*)d_in[base + 4];
    const float* g  = (const float*)d_in[base + 5];
    dim3 gd(32 / 16, NN / 16);
    linear_wmma_kernel<<<gd, 32, 0, stream>>>(emb, W1,           b1,      Abuf + (size_t)d * NN * 32, 32, 32, 32);
    linear_wmma_kernel<<<gd, 32, 0, stream>>>(emb, W1 + 32 * 32, nullptr, Bbuf + (size_t)d * NN * 32, 32, 32, 32);
    p.g[d] = g; p.bn[d] = bn; p.w2[d] = W2; p.b2[d] = b2;
  }

  // all-pairs decode: 1M pairs, LDS-tiled
  pair_decoder_kernel<<<dim3(NN / 16, NN / 16), dim3(16, 16), 0, stream>>>(
      Abuf, Bbuf, p, scores);
}